// USSPA_G_76175539962315
// MI455X (gfx1250) — compile-verified
//
#include <hip/hip_runtime.h>
#include <math.h>

// ---------------------------------------------------------------------------
// Types for CDNA5 WMMA f32 16x16x4
// ---------------------------------------------------------------------------
typedef __attribute__((ext_vector_type(2))) float v2f;
typedef __attribute__((ext_vector_type(8))) float v8f;

// ---------------------------------------------------------------------------
// GEMM: Y[b] = act(W(OxC) * X[b](CxN) + bias),  X layout (B,C,N), Y (B,O,N)
// act: 0=none, 1=relu, 2=tanh
// One wave computes a 16(M) x 128(N) strip using V_WMMA_F32_16X16X4_F32
// (8 independent accumulators per wave; A fragment reused 8x per K-step).
// Fragment layouts per CDNA5 ISA 7.12.2 (f32):
//   A (16x4): lanes 0-15 -> M=lane, VGPR0=K0,VGPR1=K1; lanes 16-31 -> K2,K3
//   B (4x16): VGPR0: lanes0-15=K0,lanes16-31=K2; VGPR1: K1 / K3
//   C/D     : VGPR r: lanes0-15 -> M=r, lanes16-31 -> M=r+8, N=lane%16
//
// K loop is split: unguarded main loop over Cfloor = C & ~3, plus one guarded
// remainder step. Guarded loads use clamped addresses * {0,1} mask multiplies
// (NOT selects) so the compiler cannot sink loads into saveexec regions —
// EXEC stays all-ones with zero mask churn around the WMMAs.
// ---------------------------------------------------------------------------
__global__ void __launch_bounds__(128)
gemm_wmma_f32(const float* __restrict__ Wm, const float* __restrict__ bias,
              const float* __restrict__ X, float* __restrict__ Y,
              int O, int C, int N, int act)
{
    const int lane = threadIdx.x & 31;
    const int wid  = threadIdx.x >> 5;
    const int lm   = lane & 15;
    const int kh   = lane >> 4;           // 0: lanes 0-15, 1: lanes 16-31
    const int b    = blockIdx.z;
    const int m0   = blockIdx.y * 64 + wid * 16;
    const int n0   = blockIdx.x * 128;
    const float* __restrict__ Xb = X + (size_t)b * C * N;
    float* __restrict__ Yb = Y + (size_t)b * O * N;

    v8f acc[8] = {};
    const int row    = m0 + lm;
    const int Cfloor = C & ~3;
    // wave-uniform interior predicate (WMMA requires EXEC all-ones; keep
    // control flow uniform)
    const bool interior = (m0 + 16 <= O) && (n0 + 128 <= N);

    if (interior) {
        const float* __restrict__ wp = Wm + (size_t)row * C + 2 * kh;
        for (int k = 0; k < Cfloor; k += 4) {
            v2f a;
            a.x = wp[k];
            a.y = wp[k + 1];
            const float* __restrict__ x0 = Xb + (size_t)(k + 2 * kh) * N + n0 + lm;
            const float* __restrict__ x1 = x0 + N;
            __builtin_prefetch(x0 + 4 * (size_t)N, 0, 1);
#pragma unroll
            for (int j = 0; j < 8; ++j) {
                v2f bf;
                bf.x = x0[j * 16];
                bf.y = x1[j * 16];
                acc[j] = __builtin_amdgcn_wmma_f32_16x16x4_f32(
                    false, a, false, bf, (short)0, acc[j], false, false);
            }
        }
        if (Cfloor < C) {               // K remainder (1..3 columns), uniform
            const int   ka  = Cfloor + 2 * kh;
            const int   ka0 = (ka     < C) ? ka       : (C - 1);
            const int   ka1 = (ka + 1 < C) ? (ka + 1) : (C - 1);
            const float km0 = (ka     < C) ? 1.f : 0.f;
            const float km1 = (ka + 1 < C) ? 1.f : 0.f;
            v2f a;
            a.x = Wm[(size_t)row * C + ka0] * km0;
            a.y = Wm[(size_t)row * C + ka1] * km1;
            const float* __restrict__ x0 = Xb + (size_t)ka0 * N + n0 + lm;
            const float* __restrict__ x1 = Xb + (size_t)ka1 * N + n0 + lm;
#pragma unroll
            for (int j = 0; j < 8; ++j) {
                v2f bf;
                bf.x = x0[j * 16] * km0;
                bf.y = x1[j * 16] * km1;
                acc[j] = __builtin_amdgcn_wmma_f32_16x16x4_f32(
                    false, a, false, bf, (short)0, acc[j], false, false);
            }
        }
    } else {
        // edge tiles (O%64, N%128 or N==1 cases): fully masked, branchless
        const int   rowc  = (row < O) ? row : (O - 1);
        const float rmask = (row < O) ? 1.f : 0.f;
        for (int k = 0; k < C; k += 4) {
            const int   ka  = k + 2 * kh;
            const int   ka0 = (ka     < C) ? ka       : (C - 1);
            const int   ka1 = (ka + 1 < C) ? (ka + 1) : (C - 1);
            const float km0 = (ka     < C) ? rmask : 0.f;
            const float km1 = (ka + 1 < C) ? rmask : 0.f;
            const float kb0 = (ka     < C) ? 1.f : 0.f;
            const float kb1 = (ka + 1 < C) ? 1.f : 0.f;
            v2f a;
            a.x = Wm[(size_t)rowc * C + ka0] * km0;
            a.y = Wm[(size_t)rowc * C + ka1] * km1;
            const float* __restrict__ xr0 = Xb + (size_t)ka0 * N;
            const float* __restrict__ xr1 = Xb + (size_t)ka1 * N;
#pragma unroll
            for (int j = 0; j < 8; ++j) {
                const int   col  = n0 + j * 16 + lm;
                const int   colc = (col < N) ? col : (N - 1);
                const float cm   = (col < N) ? 1.f : 0.f;
                v2f bf;
                bf.x = xr0[colc] * (kb0 * cm);
                bf.y = xr1[colc] * (kb1 * cm);
                acc[j] = __builtin_amdgcn_wmma_f32_16x16x4_f32(
                    false, a, false, bf, (short)0, acc[j], false, false);
            }
        }
    }

#pragma unroll
    for (int j = 0; j < 8; ++j) {
        const int col = n0 + j * 16 + lm;
        if (col >= N) continue;
#pragma unroll
        for (int r = 0; r < 8; ++r) {
            const int rr = m0 + r + 8 * kh;
            if (rr >= O) continue;
            float v = acc[j][r] + bias[rr];
            if (act == 1) v = fmaxf(v, 0.f);
            else if (act == 2) v = tanhf(v);
            Yb[(size_t)rr * N + col] = v;
        }
    }
}

// ---------------------------------------------------------------------------
// Farthest point sampling: one 1024-thread block per batch.
// Points and running min-dist live in (dynamic) LDS; block-wide argmax with
// jnp.argmax tie-breaking (lowest index among maxima).
// ---------------------------------------------------------------------------
#define FPS_T 1024
__global__ void __launch_bounds__(FPS_T)
fps_kernel(const float* __restrict__ pts, float* __restrict__ out,
           int Nin, int Nout)
{
    extern __shared__ float dyn[];
    float* sx = dyn;
    float* sy = sx + Nin;
    float* sz = sy + Nin;
    float* smind = sz + Nin;
    __shared__ float rv[FPS_T];
    __shared__ int   ri[FPS_T];
    __shared__ int   s_last;

    const int t = threadIdx.x;
    const float* P = pts + (size_t)blockIdx.x * Nin * 3;
    float* Ob = out + (size_t)blockIdx.x * Nout * 3;

    for (int i = t; i < Nin; i += FPS_T) {
        sx[i] = P[i * 3 + 0];
        sy[i] = P[i * 3 + 1];
        sz[i] = P[i * 3 + 2];
        smind[i] = 1e10f;
    }
    if (t == 0) {
        s_last = 0;
        Ob[0] = P[0]; Ob[1] = P[1]; Ob[2] = P[2];
    }
    __syncthreads();

    int last = 0;
    for (int s = 1; s < Nout; ++s) {
        const float px = sx[last], py = sy[last], pz = sz[last];
        float bv = -1.f;
        int   bi = 0x7fffffff;
        for (int i = t; i < Nin; i += FPS_T) {
            const float dx = sx[i] - px, dy = sy[i] - py, dz = sz[i] - pz;
            float m = fminf(smind[i], dx * dx + dy * dy + dz * dz);
            smind[i] = m;
            if (m > bv) { bv = m; bi = i; }   // strict > keeps first max in-stride
        }
        rv[t] = bv; ri[t] = bi;
        __syncthreads();
        for (int off = FPS_T / 2; off > 0; off >>= 1) {
            if (t < off) {
                const float ov = rv[t + off];
                const int   oi = ri[t + off];
                if (ov > rv[t] || (ov == rv[t] && oi < ri[t])) {
                    rv[t] = ov; ri[t] = oi;
                }
            }
            __syncthreads();
        }
        if (t == 0) {
            const int ix = ri[0];
            s_last = ix;
            Ob[s * 3 + 0] = sx[ix];
            Ob[s * 3 + 1] = sy[ix];
            Ob[s * 3 + 2] = sz[ix];
        }
        __syncthreads();
        last = s_last;
    }
}

// ---------------------------------------------------------------------------
// Elementwise helpers
// ---------------------------------------------------------------------------
__global__ void transpose_pts_kernel(const float* __restrict__ in,
                                     float* __restrict__ out, int B, int N)
{
    const size_t t = (size_t)blockIdx.x * blockDim.x + threadIdx.x;
    const size_t total = (size_t)B * N * 3;
    if (t >= total) return;
    const int d = (int)(t % 3);
    const size_t bn = t / 3;
    const int i = (int)(bn % N);
    const int b = (int)(bn / N);
    out[((size_t)b * 3 + d) * N + i] = in[t];
}

// dst[b, cOff+c, n] = src[b, c, n]   (src is (B,Cs,N), dst has Cd channels)
__global__ void copy_chan_kernel(float* __restrict__ dst, int Cd, int cOff,
                                 const float* __restrict__ src, int Cs, int N, int B)
{
    const size_t t = (size_t)blockIdx.x * blockDim.x + threadIdx.x;
    const size_t total = (size_t)B * Cs * N;
    if (t >= total) return;
    const int n = (int)(t % N);
    const size_t bc = t / N;
    const int c = (int)(bc % Cs);
    const int b = (int)(bc / Cs);
    dst[((size_t)b * Cd + cOff + c) * N + n] = src[t];
}

// dst[b, cOff+c, n] = src[b, c]  (broadcast along N)
__global__ void bcast_chan_kernel(float* __restrict__ dst, int Cd, int cOff,
                                  const float* __restrict__ src, int Cs, int N, int B)
{
    const size_t t = (size_t)blockIdx.x * blockDim.x + threadIdx.x;
    const size_t total = (size_t)B * Cs * N;
    if (t >= total) return;
    const int n = (int)(t % N);
    const size_t bc = t / N;
    const int c = (int)(bc % Cs);
    const int b = (int)(bc / Cs);
    dst[((size_t)b * Cd + cOff + c) * N + n] = src[(size_t)b * Cs + c];
}

// dst (B,Nd,3); dst[b, rOff+i, :] = src[b, i, :]  (src is (B,Ns,3))
__global__ void cat_rows_kernel(float* __restrict__ dst, int Nd, int rOff,
                                const float* __restrict__ src, int Ns, int B)
{
    const size_t t = (size_t)blockIdx.x * blockDim.x + threadIdx.x;
    const size_t total = (size_t)B * Ns * 3;
    if (t >= total) return;
    const int d = (int)(t % 3);
    const size_t bn = t / 3;
    const int i = (int)(bn % Ns);
    const int b = (int)(bn / Ns);
    dst[((size_t)b * Nd + rOff + i) * 3 + d] = src[t];
}

// rowmax over N: out[row] = max_n X[row, n], row = b*C + c
__global__ void __launch_bounds__(128)
rowmax_kernel(const float* __restrict__ X, float* __restrict__ out, int N)
{
    __shared__ float red[128];
    const int row = blockIdx.x;
    const float* xr = X + (size_t)row * N;
    float m = -3.4e38f;
    for (int i = threadIdx.x; i < N; i += 128) m = fmaxf(m, xr[i]);
    red[threadIdx.x] = m;
    __syncthreads();
    for (int off = 64; off > 0; off >>= 1) {
        if (threadIdx.x < off)
            red[threadIdx.x] = fmaxf(red[threadIdx.x], red[threadIdx.x + off]);
        __syncthreads();
    }
    if (threadIdx.x == 0) out[row] = red[0];
}

// get_mirror: per (b,i) reflect across plane (A, 0, Bc)
__global__ void mirror_kernel(const float* __restrict__ pts,
                              const float* __restrict__ ab,
                              float* __restrict__ out, int B, int N)
{
    const size_t t = (size_t)blockIdx.x * blockDim.x + threadIdx.x;
    if (t >= (size_t)B * N) return;
    const int b = (int)(t / N);
    const float A  = ab[b * 2 + 0];
    const float Bc = ab[b * 2 + 1];
    const float* p = pts + t * 3;
    const float x = p[0], y = p[1], z = p[2];
    const float ax = 2.f * (A * x + Bc * z) / (A * A + Bc * Bc + 1e-8f);
    float* o = out + t * 3;
    o[0] = x - A * ax;
    o[1] = y;
    o[2] = z - Bc * ax;
}

// ball_group index selection: first ns indices j (ascending) with d2 < r2,
// remainder filled with the first valid index (matches argsort-on-masked-key).
__global__ void ball_idx_kernel(const float* __restrict__ pts,
                                int* __restrict__ idx,
                                int B, int N, float r2, int ns)
{
    const size_t t = (size_t)blockIdx.x * blockDim.x + threadIdx.x;
    if (t >= (size_t)B * N) return;
    const int b = (int)(t / N);
    const int i = (int)(t - (size_t)b * N);
    const float* P = pts + (size_t)b * N * 3;
    const float xi = P[i * 3], yi = P[i * 3 + 1], zi = P[i * 3 + 2];
    int* row = idx + t * ns;
    int cnt = 0, first = -1;
    for (int j = 0; j < N && cnt < ns; ++j) {
        const float dx = P[j * 3] - xi;
        const float dy = P[j * 3 + 1] - yi;
        const float dz = P[j * 3 + 2] - zi;
        if (dx * dx + dy * dy + dz * dz < r2) {
            if (first < 0) first = j;
            row[cnt++] = j;
        }
    }
    if (first < 0) first = i;  // unreachable (self distance 0), but safe
    for (int s = cnt; s < ns; ++s) row[s] = first;
}

// out[b,c,i] = max_s x[b,c, idx[b,i,s]]
__global__ void group_max_kernel(const float* __restrict__ x,
                                 const int* __restrict__ idx,
                                 float* __restrict__ out,
                                 int B, int Cc, int N, int ns)
{
    const size_t t = (size_t)blockIdx.x * blockDim.x + threadIdx.x;
    const size_t total = (size_t)B * Cc * N;
    if (t >= total) return;
    const int i = (int)(t % N);
    const size_t bc = t / N;
    const int c = (int)(bc % Cc);
    const int b = (int)(bc / Cc);
    const float* xb = x + ((size_t)b * Cc + c) * N;
    const int* row = idx + ((size_t)b * N + i) * ns;
    float m = -3.4e38f;
    for (int s = 0; s < ns; ++s) m = fmaxf(m, xb[row[s]]);
    out[t] = m;
}

// out[b, i*4+u, d] = pts[b, i, d] + shift[b, u*3+d, i]
// pts (B,512,3), shift (B,12,512), out (B,2048,3)
__global__ void add_shift_kernel(const float* __restrict__ pts,
                                 const float* __restrict__ sh,
                                 float* __restrict__ out, int B)
{
    const size_t t = (size_t)blockIdx.x * blockDim.x + threadIdx.x;
    const size_t total = (size_t)B * 2048 * 3;
    if (t >= total) return;
    const int d = (int)(t % 3);
    const int r = (int)((t / 3) % 2048);
    const int b = (int)(t / (3 * 2048));
    const int i = r >> 2;
    const int u = r & 3;
    out[t] = pts[((size_t)b * 512 + i) * 3 + d] +
             sh[((size_t)b * 12 + (u * 3 + d)) * 512 + i];
}

// ---------------------------------------------------------------------------
// Host-side helpers
// ---------------------------------------------------------------------------
static inline int cdiv(int a, int b) { return (a + b - 1) / b; }

static inline void gemm(const float* W, const float* b, const float* X, float* Y,
                        int O, int C, int N, int B, int act, hipStream_t s)
{
    dim3 g(cdiv(N, 128), cdiv(O, 64), B);
    gemm_wmma_f32<<<g, dim3(128), 0, s>>>(W, b, X, Y, O, C, N, act);
}

#define EW_GRID(total) dim3((unsigned)(((total) + 255) / 256)), dim3(256)

// pcn_encoder: xT (B,3,2048) -> f (B,512). Uses BUFA/BUFB ping-pong and G.
static void run_pcn_encoder(const float* xT, int B,
                            const float* W0, const float* b0,
                            const float* W1, const float* b1,
                            const float* W2, const float* b2,
                            const float* W3, const float* b3,
                            float* BUFA, float* BUFB, float* G, float* fout,
                            hipStream_t s)
{
    const int N = 2048;
    gemm(W0, b0, xT,   BUFA, 128, 3,   N, B, 1, s);
    gemm(W1, b1, BUFA, BUFB, 256, 128, N, B, 0, s);
    rowmax_kernel<<<dim3(B * 256), dim3(128), 0, s>>>(BUFB, G, N);
    {
        size_t tot = (size_t)B * 256 * N;
        copy_chan_kernel<<<EW_GRID(tot), 0, s>>>(BUFA, 512, 0, BUFB, 256, N, B);
        bcast_chan_kernel<<<EW_GRID(tot), 0, s>>>(BUFA, 512, 256, G, 256, N, B);
    }
    gemm(W2, b2, BUFA, BUFB, 512, 512, N, B, 1, s);
    gemm(W3, b3, BUFB, BUFA, 512, 512, N, B, 0, s);
    rowmax_kernel<<<dim3(B * 512), dim3(128), 0, s>>>(BUFA, fout, N);
}

// ---------------------------------------------------------------------------
// Entry point
// ---------------------------------------------------------------------------
extern "C" void kernel_launch(void* const* d_in, const int* in_sizes, int n_in,
                              void* d_out, int out_size, void* d_ws, size_t ws_size,
                              hipStream_t stream)
{
    (void)in_sizes; (void)n_in; (void)out_size; (void)ws_size;
    const float* inR = (const float*)d_in[0];   // (16,2048,3)
    const float* inA = (const float*)d_in[1];   // (16,2048,3)
#define P(i) ((const float*)d_in[(i)])

    // ---------------- workspace carve-out ----------------
    char* wsp = (char*)d_ws;
    auto alloc = [&](size_t nfloats) -> float* {
        float* p = (float*)wsp;
        wsp += ((nfloats * sizeof(float) + 255) & ~(size_t)255);
        return p;
    };
    float* XT     = alloc((size_t)32 * 3 * 2048);
    float* BUFA   = alloc((size_t)32 * 512 * 2048);
    float* BUFB   = alloc((size_t)32 * 512 * 2048);
    float* G      = alloc((size_t)32 * 512);
    float* F      = alloc((size_t)32 * 512);
    float* FR0    = alloc((size_t)16 * 512);
    float* S0     = alloc((size_t)32 * 1024);
    float* S1     = alloc((size_t)32 * 1024);
    float* DECR   = alloc((size_t)16 * 1536);
    float* DEC    = alloc((size_t)32 * 1536);
    float* AB     = alloc(32);
    float* CAT1   = alloc((size_t)16 * 4608 * 3);
    float* PTS32  = alloc((size_t)32 * 2048 * 3);
    float* RT     = alloc((size_t)32 * 3 * 512);
    float* SH     = alloc((size_t)32 * 12 * 512);
    float* REFOUT = alloc((size_t)32 * 2048 * 3);
    float* PM     = alloc((size_t)16 * 2048 * 3);
    float* CAT2   = alloc((size_t)16 * 4096 * 3);
    int*   IDXB   = (int*)alloc((size_t)32 * 512 * 32);
    // refine buffers aliased into the (now-idle) encoder ping/pong buffers
    const size_t RSEG = (size_t)32 * 512 * 512;
    float* RA    = BUFA;
    float* RB    = BUFA + RSEG;
    float* RC512 = BUFA + 2 * RSEG;
    float* RC515 = BUFB;               // 32*515*512 < |BUFB|

    // ---------------- output slots (return-tuple order) ----------------
    float* out = (float*)d_out;
    float* o_fR   = out;                 //  16*512        = 8192
    float* o_fA   = out + 8192;          //  16*512
    float* o_pR   = out + 16384;         //  16*512*3      = 24576
    float* o_pR3  = out + 40960;         //  16*2048*3     = 98304
    float* o_pA   = out + 139264;        //  16*512*3
    float* o_pA3  = out + 163840;        //  16*2048*3
    float* o_inRp = out + 262144;        //  16*2048*3
    float* o_pR0  = out + 360448;        //  16*512*3
    float* o_inRM = out + 385024;        //  16*2048*3  (total 483328)

    // ================= Stage 1: E_R encoder =================
    {
        size_t tot = (size_t)16 * 2048 * 3;
        transpose_pts_kernel<<<EW_GRID(tot), 0, stream>>>(inR, XT, 16, 2048);
    }
    run_pcn_encoder(XT, 16, P(2), P(3), P(4), P(5), P(6), P(7), P(8), P(9),
                    BUFA, BUFB, G, FR0, stream);

    // ================= Stage 2: D_R decoder (N=1) =================
    gemm(P(18), P(19), FR0, S0, 512, 512, 1, 16, 1, stream);
    gemm(P(20), P(21), S0, S1, 512, 512, 1, 16, 1, stream);
    gemm(P(22), P(23), S1, S0, 1024, 512, 1, 16, 1, stream);
    gemm(P(24), P(25), S0, S1, 1024, 1024, 1, 16, 1, stream);
    gemm(P(26), P(27), S1, DECR, 1536, 1024, 1, 16, 0, stream);
    hipMemcpyAsync(o_pR0, DECR, (size_t)16 * 1536 * 4, hipMemcpyDeviceToDevice, stream);

    // ================= Stage 3: ab head (tanh) =================
    gemm(P(38), P(39), FR0, S0, 128, 512, 1, 16, 1, stream);
    gemm(P(40), P(41), S0, S1, 128, 128, 1, 16, 1, stream);
    gemm(P(42), P(43), S1, AB, 2, 128, 1, 16, 2 /*tanh*/, stream);

    // ================= Stage 4: mirror of input_R =================
    mirror_kernel<<<EW_GRID((size_t)16 * 2048), 0, stream>>>(inR, AB, o_inRM, 16, 2048);

    // ================= Stage 5: cat + first FPS =================
    cat_rows_kernel<<<EW_GRID((size_t)16 * 2048 * 3), 0, stream>>>(CAT1, 4608, 0, inR, 2048, 16);
    cat_rows_kernel<<<EW_GRID((size_t)16 * 2048 * 3), 0, stream>>>(CAT1, 4608, 2048, o_inRM, 2048, 16);
    cat_rows_kernel<<<EW_GRID((size_t)16 * 512 * 3), 0, stream>>>(CAT1, 4608, 4096, DECR, 512, 16);
    fps_kernel<<<dim3(16), dim3(FPS_T), 4 * 4608 * sizeof(float), stream>>>(CAT1, o_inRp, 4608, 2048);

    // ================= Stage 6: E_A encoder on [fps_out; input_A] =================
    hipMemcpyAsync(PTS32, o_inRp, (size_t)16 * 2048 * 3 * 4, hipMemcpyDeviceToDevice, stream);
    hipMemcpyAsync(PTS32 + (size_t)16 * 2048 * 3, inA, (size_t)16 * 2048 * 3 * 4,
                   hipMemcpyDeviceToDevice, stream);
    transpose_pts_kernel<<<EW_GRID((size_t)32 * 2048 * 3), 0, stream>>>(PTS32, XT, 32, 2048);
    run_pcn_encoder(XT, 32, P(10), P(11), P(12), P(13), P(14), P(15), P(16), P(17),
                    BUFA, BUFB, G, F, stream);
    hipMemcpyAsync(o_fR, F, (size_t)16 * 512 * 4, hipMemcpyDeviceToDevice, stream);
    hipMemcpyAsync(o_fA, F + (size_t)16 * 512, (size_t)16 * 512 * 4, hipMemcpyDeviceToDevice, stream);

    // ================= Stage 7: D_A decoder (N=1, B=32) =================
    gemm(P(28), P(29), F, S0, 512, 512, 1, 32, 1, stream);
    gemm(P(30), P(31), S0, S1, 512, 512, 1, 32, 1, stream);
    gemm(P(32), P(33), S1, S0, 1024, 512, 1, 32, 1, stream);
    gemm(P(34), P(35), S0, S1, 1024, 1024, 1, 32, 1, stream);
    gemm(P(36), P(37), S1, DEC, 1536, 1024, 1, 32, 0, stream);
    hipMemcpyAsync(o_pR, DEC, (size_t)16 * 1536 * 4, hipMemcpyDeviceToDevice, stream);
    hipMemcpyAsync(o_pA, DEC + (size_t)16 * 1536, (size_t)16 * 1536 * 4,
                   hipMemcpyDeviceToDevice, stream);

    // ================= Stage 8: upsampling_refine (B=32, N=512) =================
    const int RN = 512, RB32 = 32;
    transpose_pts_kernel<<<EW_GRID((size_t)32 * 512 * 3), 0, stream>>>(DEC, RT, 32, 512);
    // r1
    gemm(P(44), P(45), RT, RA, 256, 3, RN, RB32, 1, stream);
    gemm(P(46), P(47), RA, RB, 256, 256, RN, RB32, 1, stream);
    gemm(P(48), P(49), RB, RA, 256, 256, RN, RB32, 0, stream);
    rowmax_kernel<<<dim3(32 * 256), dim3(128), 0, stream>>>(RA, G, RN);
    {
        size_t tot = (size_t)32 * 256 * RN;
        copy_chan_kernel<<<EW_GRID(tot), 0, stream>>>(RC512, 512, 0, RA, 256, RN, RB32);
        bcast_chan_kernel<<<EW_GRID(tot), 0, stream>>>(RC512, 512, 256, G, 256, RN, RB32);
    }
    // r2
    gemm(P(50), P(51), RC512, RA, 256, 512, RN, RB32, 1, stream);
    gemm(P(52), P(53), RA, RB, 256, 256, RN, RB32, 1, stream);
    gemm(P(54), P(55), RB, RA, 256, 256, RN, RB32, 0, stream);
    // ball grouping + local max
    ball_idx_kernel<<<EW_GRID((size_t)32 * 512), 0, stream>>>(DEC, IDXB, 32, 512, 0.0625f, 32);
    group_max_kernel<<<EW_GRID((size_t)32 * 256 * 512), 0, stream>>>(RA, IDXB, RB, 32, 256, 512, 32);
    {
        size_t tot = (size_t)32 * 256 * RN;
        copy_chan_kernel<<<EW_GRID(tot), 0, stream>>>(RC512, 512, 0, RA, 256, RN, RB32);
        copy_chan_kernel<<<EW_GRID(tot), 0, stream>>>(RC512, 512, 256, RB, 256, RN, RB32);
    }
    // r3
    gemm(P(56), P(57), RC512, RA, 256, 512, RN, RB32, 1, stream);
    gemm(P(58), P(59), RA, RB, 256, 256, RN, RB32, 1, stream);
    gemm(P(60), P(61), RB, RA, 256, 256, RN, RB32, 0, stream);
    rowmax_kernel<<<dim3(32 * 256), dim3(128), 0, stream>>>(RA, G, RN);
    // concat [pointT(3), x(256), bcast xm(256)] -> (32,515,512)
    copy_chan_kernel<<<EW_GRID((size_t)32 * 3 * RN), 0, stream>>>(RC515, 515, 0, RT, 3, RN, RB32);
    copy_chan_kernel<<<EW_GRID((size_t)32 * 256 * RN), 0, stream>>>(RC515, 515, 3, RA, 256, RN, RB32);
    bcast_chan_kernel<<<EW_GRID((size_t)32 * 256 * RN), 0, stream>>>(RC515, 515, 259, G, 256, RN, RB32);
    // r4
    gemm(P(62), P(63), RC515, RB, 512, 515, RN, RB32, 1, stream);
    gemm(P(64), P(65), RB, RA, 512, 512, RN, RB32, 1, stream);
    gemm(P(66), P(67), RA, SH, 12, 512, RN, RB32, 0, stream);
    add_shift_kernel<<<EW_GRID((size_t)32 * 2048 * 3), 0, stream>>>(DEC, SH, REFOUT, 32);
    hipMemcpyAsync(o_pA3, REFOUT + (size_t)16 * 2048 * 3, (size_t)16 * 2048 * 3 * 4,
                   hipMemcpyDeviceToDevice, stream);

    // ================= Stage 9: mirror + final FPS =================
    mirror_kernel<<<EW_GRID((size_t)16 * 2048), 0, stream>>>(REFOUT, AB, PM, 16, 2048);
    cat_rows_kernel<<<EW_GRID((size_t)16 * 2048 * 3), 0, stream>>>(CAT2, 4096, 0, REFOUT, 2048, 16);
    cat_rows_kernel<<<EW_GRID((size_t)16 * 2048 * 3), 0, stream>>>(CAT2, 4096, 2048, PM, 2048, 16);
    fps_kernel<<<dim3(16), dim3(FPS_T), 4 * 4096 * sizeof(float), stream>>>(CAT2, o_pR3, 4096, 2048);
#undef P
}